// EdgeUpdateLayer_34376918237285
// MI455X (gfx1250) — compile-verified
//
#include <hip/hip_runtime.h>

// ---------------- problem constants ----------------
#define NODE_DIM 128
#define EDGE_DIM 64
#define HID      128
#define IN_DIM   320              // 2*NODE_DIM + EDGE_DIM
#define K1C      (IN_DIM / 4)     // 80  K-chunks of 4 for GEMM1
#define K2C      (HID / 4)        // 32  K-chunks of 4 for GEMM2
#define WAVES    8
#define TPB      (WAVES * 32)
#define NWG      512

typedef float v2f __attribute__((ext_vector_type(2)));
typedef float v8f __attribute__((ext_vector_type(8)));

// hs per-wave scratch: per K2-chunk, 16 rows (float4) + 1 pad float4 -> 68 dwords
#define HS_STRIDE 17              // float4 per k2-chunk
#define HS_SIZE   (K2C * HS_STRIDE)   // float4 per wave = 544 (8704 B)

__global__ __launch_bounds__(TPB) void edge_mlp_wmma(
    const float* __restrict__ node,  const float* __restrict__ edgef,
    const int*   __restrict__ eidx,  const float* __restrict__ W1,
    const float* __restrict__ b1,    const float* __restrict__ W2,
    const float* __restrict__ b2,    float* __restrict__ out,
    int E, int nblocks)
{
    // ---- LDS: weights swizzled so a B-fragment half is one contiguous float2 ----
    // W1s[k*HID + n] = (W1[4k+0][n], W1[4k+1][n], W1[4k+2][n], W1[4k+3][n])
    __shared__ float4 W1s[K1C * HID];        // 160 KB
    __shared__ float4 W2s[K2C * EDGE_DIM];   //  32 KB
    __shared__ float4 hs[WAVES][HS_SIZE];    //  68 KB  (per-wave h tile, A2-swizzled)
    __shared__ float  b1s[HID];
    __shared__ float  b2s[EDGE_DIM];

    const int tid = threadIdx.x;

    for (int i = tid; i < K1C * HID; i += TPB) {
        const int k = i / HID, n = i - k * HID;
        const float* p = W1 + (4 * k) * HID + n;
        W1s[i] = make_float4(p[0], p[HID], p[2 * HID], p[3 * HID]);
    }
    for (int i = tid; i < K2C * EDGE_DIM; i += TPB) {
        const int k = i / EDGE_DIM, n = i - k * EDGE_DIM;
        const float* p = W2 + (4 * k) * EDGE_DIM + n;
        W2s[i] = make_float4(p[0], p[EDGE_DIM], p[2 * EDGE_DIM], p[3 * EDGE_DIM]);
    }
    if (tid < HID)      b1s[tid] = b1[tid];
    if (tid < EDGE_DIM) b2s[tid] = b2[tid];
    __syncthreads();

    const int lane = tid & 31;
    const int wv   = tid >> 5;
    const int m    = lane & 15;   // M row (edge) / N column within a 16-wide tile
    const int hi   = lane >> 4;   // half-wave: selects K pair {0,1} vs {2,3}

    float* __restrict__ hsw = (float*)&hs[wv][0];

    const int wave_id = blockIdx.x * WAVES + wv;
    const int wstride = gridDim.x * WAVES;

    for (int blk = wave_id; blk < nblocks; blk += wstride) {
        // prefetch next block's edge row + indices (global_prefetch_b8)
        const int nblk = blk + wstride;
        if (nblk < nblocks) {
            __builtin_prefetch(edgef + (long)(nblk * 16 + m) * EDGE_DIM, 0, 0);
            __builtin_prefetch(eidx + nblk * 16 + m, 0, 0);
        }

        const int e = blk * 16 + m;
        const int s = eidx[e];
        const int d = eidx[E + e];
        const float* __restrict__ rowS = node  + (long)s * NODE_DIM;
        const float* __restrict__ rowD = node  + (long)d * NODE_DIM;
        const float* __restrict__ rowE = edgef + (long)e * EDGE_DIM;

        // ---------------- GEMM1: (16 x 320) @ (320 x 128) ----------------
        v8f acc[8] = {};
        for (int k = 0; k < 32; ++k) {            // K columns [0,128): node[src]
            v2f a = *(const v2f*)(rowS + 4 * k + 2 * hi);
            #pragma unroll
            for (int j = 0; j < 8; ++j) {
                v2f b = *((const v2f*)&W1s[k * HID + 16 * j + m] + hi);
                acc[j] = __builtin_amdgcn_wmma_f32_16x16x4_f32(
                    false, a, false, b, (short)0, acc[j], false, false);
            }
        }
        for (int k = 0; k < 32; ++k) {            // K columns [128,256): node[dst]
            v2f a = *(const v2f*)(rowD + 4 * k + 2 * hi);
            #pragma unroll
            for (int j = 0; j < 8; ++j) {
                v2f b = *((const v2f*)&W1s[(k + 32) * HID + 16 * j + m] + hi);
                acc[j] = __builtin_amdgcn_wmma_f32_16x16x4_f32(
                    false, a, false, b, (short)0, acc[j], false, false);
            }
        }
        for (int k = 0; k < 16; ++k) {            // K columns [256,320): edge feats
            v2f a = *(const v2f*)(rowE + 4 * k + 2 * hi);
            #pragma unroll
            for (int j = 0; j < 8; ++j) {
                v2f b = *((const v2f*)&W1s[(k + 64) * HID + 16 * j + m] + hi);
                acc[j] = __builtin_amdgcn_wmma_f32_16x16x4_f32(
                    false, a, false, b, (short)0, acc[j], false, false);
            }
        }

        // ---- bias + ReLU, spill h to LDS pre-swizzled into A-fragment order ----
        // C/D layout: acc[j][r] at this lane = h[row = r + 8*hi][col = 16*j + m]
        #pragma unroll
        for (int j = 0; j < 8; ++j) {
            const int c  = 16 * j + m;            // hidden column = GEMM2 K index
            const float bias = b1s[c];
            const int k2 = c >> 2, pos = c & 3;
            #pragma unroll
            for (int r = 0; r < 8; ++r) {
                float v = acc[j][r] + bias;
                v = v > 0.0f ? v : 0.0f;
                const int row = r + 8 * hi;       // edge row
                hsw[(k2 * HS_STRIDE + row) * 4 + pos] = v;
            }
        }

        // ---------------- GEMM2: (16 x 128) @ (128 x 64) ----------------
        v8f acc2[4] = {};
        for (int k2 = 0; k2 < K2C; ++k2) {
            v2f a2 = *(const v2f*)(hsw + k2 * (HS_STRIDE * 4) + m * 4 + hi * 2);
            #pragma unroll
            for (int j2 = 0; j2 < 4; ++j2) {
                v2f b = *((const v2f*)&W2s[k2 * EDGE_DIM + 16 * j2 + m] + hi);
                acc2[j2] = __builtin_amdgcn_wmma_f32_16x16x4_f32(
                    false, a2, false, b, (short)0, acc2[j2], false, false);
            }
        }

        // ---------------- bias + store ----------------
        #pragma unroll
        for (int j2 = 0; j2 < 4; ++j2) {
            const float bias = b2s[16 * j2 + m];
            #pragma unroll
            for (int r = 0; r < 8; ++r) {
                const int row = r + 8 * hi;
                out[(long)(blk * 16 + row) * EDGE_DIM + 16 * j2 + m] =
                    acc2[j2][r] + bias;
            }
        }
    }
}

// Scalar fallback for a ragged tail (unused when E % 16 == 0, as here).
__global__ void edge_mlp_tail(
    const float* __restrict__ node,  const float* __restrict__ edgef,
    const int*   __restrict__ eidx,  const float* __restrict__ W1,
    const float* __restrict__ b1,    const float* __restrict__ W2,
    const float* __restrict__ b2,    float* __restrict__ out,
    int E, int start)
{
    const int e = start + blockIdx.x;
    if (e >= E) return;
    const int t = threadIdx.x;  // 64 threads, one output column each
    __shared__ float comb[IN_DIM];
    __shared__ float hsh[HID];
    const int s = eidx[e], d = eidx[E + e];
    for (int i = t; i < NODE_DIM; i += 64) {
        comb[i]            = node[(long)s * NODE_DIM + i];
        comb[NODE_DIM + i] = node[(long)d * NODE_DIM + i];
    }
    if (t < EDGE_DIM) comb[2 * NODE_DIM + t] = edgef[(long)e * EDGE_DIM + t];
    __syncthreads();
    for (int hcol = t; hcol < HID; hcol += 64) {
        float a = b1[hcol];
        for (int k = 0; k < IN_DIM; ++k) a += comb[k] * W1[k * HID + hcol];
        hsh[hcol] = a > 0.0f ? a : 0.0f;
    }
    __syncthreads();
    float a = b2[t];
    for (int k = 0; k < HID; ++k) a += hsh[k] * W2[k * EDGE_DIM + t];
    out[(long)e * EDGE_DIM + t] = a;
}

extern "C" void kernel_launch(void* const* d_in, const int* in_sizes, int n_in,
                              void* d_out, int out_size, void* d_ws, size_t ws_size,
                              hipStream_t stream) {
    const float* node  = (const float*)d_in[0];
    const float* edgef = (const float*)d_in[1];
    const int*   eidx  = (const int*)  d_in[2];
    const float* W1    = (const float*)d_in[3];
    const float* b1    = (const float*)d_in[4];
    const float* W2    = (const float*)d_in[5];
    const float* b2    = (const float*)d_in[6];
    float* out = (float*)d_out;

    const int E = in_sizes[1] / EDGE_DIM;   // 1,600,000
    const int nblocks = E / 16;             // 100,000 (exact)

    int grid = NWG;
    const int need = (nblocks + WAVES - 1) / WAVES;
    if (grid > need) grid = need > 0 ? need : 1;

    if (nblocks > 0)
        edge_mlp_wmma<<<grid, TPB, 0, stream>>>(node, edgef, eidx, W1, b1, W2, b2,
                                                out, E, nblocks);
    const int tail = E - nblocks * 16;
    if (tail > 0)
        edge_mlp_tail<<<tail, 64, 0, stream>>>(node, edgef, eidx, W1, b1, W2, b2,
                                               out, E, nblocks * 16);
}